// crossAttention_2765958939220
// MI455X (gfx1250) — compile-verified
//
#include <hip/hip_runtime.h>
#include <math.h>

// Problem constants (from reference): B=2, C=1024, I=128, D=64
#define PB 2
#define PC 1024
#define PI 128
#define PD 64

typedef float v2f __attribute__((ext_vector_type(2)));
typedef float v8f __attribute__((ext_vector_type(8)));

// ---------------------------------------------------------------------------
// Kernel 1: camsum[b,c,d] = sum_i cam[b,c,i,d]
// One block per (b,c); 256 threads = 4 row-groups x 64 d-lanes.
// Each i-step: 64 lanes load 256 contiguous bytes -> fully coalesced.
// ---------------------------------------------------------------------------
__global__ void reduce_i_kernel(const float* __restrict__ cam,
                                float* __restrict__ camsum) {
    __shared__ float sm[256];
    const int bc = blockIdx.x;                    // 0 .. B*C-1
    const float* src = cam + (size_t)bc * (PI * PD);
    const int t = threadIdx.x;
    const int d = t & 63;
    const int g = t >> 6;                          // row group 0..3
    float s = 0.0f;
#pragma unroll 8
    for (int i = g; i < PI; i += 4) s += src[i * PD + d];
    sm[t] = s;
    __syncthreads();
    if (g == 0) {
        camsum[(size_t)bc * PD + d] = sm[d] + sm[64 + d] + sm[128 + d] + sm[192 + d];
    }
}

// ---------------------------------------------------------------------------
// Kernel 2/3: Y[b] = W (CxC, row-major [o][c]) x X[b] (C x D, [b][c][d])
// One wave (32 threads) per 16x16 output tile, K-loop in steps of 4 using
// V_WMMA_F32_16X16X4_F32.
//
// Fragment layouts (ISA 7.12.2, 32-bit):
//   A 16x4 : lanes 0-15 hold M=0..15; VGPR0 = K-even-pair lo, VGPR1 = next.
//            lanes 0-15: {K,K+1}; lanes 16-31: {K+2,K+3}
//   B 4x16 : one row striped across lanes in a VGPR, mirrored split:
//            VGPR0: row K (lanes 0-15) / row K+2 (lanes 16-31); VGPR1: K+1/K+3
//   C/D    : 8 VGPRs; VGPR r: lanes 0-15 -> M=r, lanes 16-31 -> M=r+8; N=lane&15
// ---------------------------------------------------------------------------
__global__ void gemm_wmma_f32(const float* __restrict__ W,
                              const float* __restrict__ X,
                              float* __restrict__ Y) {
    const int id = blockIdx.x;
    const int nt = id % (PD / 16);                 // N-tile: 0..3
    const int mt = (id / (PD / 16)) % (PC / 16);   // M-tile: 0..63
    const int b  = id / ((PD / 16) * (PC / 16));   // batch
    const int lane = threadIdx.x;                  // 0..31 (wave32)
    const int half = lane >> 4;                    // 0 | 1
    const int l15  = lane & 15;

    const float* Wp = W + (size_t)(mt * 16 + l15) * PC;          // row o
    const float* Xp = X + (size_t)b * PC * PD + nt * 16 + l15;   // col d

    v8f acc = {};
    for (int k = 0; k < PC; k += 4) {
        const int kk = k + 2 * half;
        v2f a;                                    // A: W[o][kk], W[o][kk+1]
        a.x = Wp[kk + 0];
        a.y = Wp[kk + 1];
        v2f bm;                                   // B: X[kk][d], X[kk+1][d]
        bm.x = Xp[(size_t)(kk + 0) * PD];
        bm.y = Xp[(size_t)(kk + 1) * PD];
        acc = __builtin_amdgcn_wmma_f32_16x16x4_f32(
            /*neg_a=*/false, a, /*neg_b=*/false, bm,
            /*c_mod=*/(short)0, acc, /*reuse_a=*/false, /*reuse_b=*/false);
    }

    float* Yp = Y + (size_t)b * PC * PD + (size_t)(mt * 16) * PD + nt * 16 + l15;
#pragma unroll
    for (int r = 0; r < 8; ++r) {
        Yp[(size_t)(r + 8 * half) * PD] = acc[r];
    }
}

// ---------------------------------------------------------------------------
// Kernel 4: out[b,c,i,d] = gelu_exact(y[b,c,d]) broadcast over i (write 64 MiB)
// One block per (b,c); threads own a float4 of the 64-float row; 16 rows per
// store pass (lanes 0-15 cover one contiguous 256B row).
// ---------------------------------------------------------------------------
__device__ __forceinline__ float gelu_exact(float x) {
    return 0.5f * x * (1.0f + erff(x * 0.70710678118654752440f));
}

__global__ void gelu_broadcast_kernel(const float* __restrict__ y,
                                      float* __restrict__ out) {
    const int bc = blockIdx.x;
    const int t = threadIdx.x;           // 0..255
    const int d4 = t & 15;               // float4 index within row
    const int r  = t >> 4;               // 0..15 row group
    const float4* yrow = (const float4*)(y + (size_t)bc * PD);
    float4 v = yrow[d4];
    v.x = gelu_exact(v.x);
    v.y = gelu_exact(v.y);
    v.z = gelu_exact(v.z);
    v.w = gelu_exact(v.w);
    float4* dst = (float4*)(out + (size_t)bc * (size_t)(PI * PD));
#pragma unroll
    for (int i = r; i < PI; i += 16) {
        dst[i * (PD / 4) + d4] = v;
    }
}

// ---------------------------------------------------------------------------
// Launch: inputs are [lidar, cam, WQ, WK, WV, Wout]; lidar/WQ/WK are dead
// (softmax rows sum to 1 => attention output == broadcast column-sums of v).
// ---------------------------------------------------------------------------
extern "C" void kernel_launch(void* const* d_in, const int* in_sizes, int n_in,
                              void* d_out, int out_size, void* d_ws, size_t ws_size,
                              hipStream_t stream) {
    (void)in_sizes; (void)n_in; (void)out_size; (void)ws_size;
    const float* cam  = (const float*)d_in[1];
    const float* WV   = (const float*)d_in[4];
    const float* Wout = (const float*)d_in[5];
    float* out = (float*)d_out;

    float* camsum = (float*)d_ws;                 // B*C*D floats (512 KiB)
    float* tbuf   = camsum + (size_t)PB * PC * PD;
    float* ybuf   = tbuf   + (size_t)PB * PC * PD;

    // 1) reduce cam over I
    reduce_i_kernel<<<PB * PC, 256, 0, stream>>>(cam, camsum);
    // 2) t = WV @ camsum   (per batch, 1024x1024 x 1024x64)
    const int gemm_blocks = PB * (PC / 16) * (PD / 16);
    gemm_wmma_f32<<<gemm_blocks, 32, 0, stream>>>(WV, camsum, tbuf);
    // 3) y = Wout @ t
    gemm_wmma_f32<<<gemm_blocks, 32, 0, stream>>>(Wout, tbuf, ybuf);
    // 4) out = gelu(y) broadcast over i
    gelu_broadcast_kernel<<<PB * PC, 256, 0, stream>>>(ybuf, out);
}